// ExtendedKalmanVAE_84370337563325
// MI455X (gfx1250) — compile-verified
//
#include <hip/hip_runtime.h>

// Extended Kalman VAE filter+smoother for MI455X (gfx1250, wave32).
// One wave32 per batch element; all 32x32xK matmuls via V_WMMA_F32_16X16X4_F32.
// Weights staged into LDS via the Tensor Data Mover (tensor_load_to_lds).

#define LATENT 32
#define OBSN   32
#define HIDN   128
#define BATCH  256
#define TSTEPS 100
#define WPB    2              // waves (batch elements) per block
#define NTHREADS (WPB * 32)

typedef __attribute__((ext_vector_type(2))) float v2f;
typedef __attribute__((ext_vector_type(8))) float v8f;
typedef __attribute__((ext_vector_type(4))) unsigned int v4u;
typedef __attribute__((ext_vector_type(4))) int v4i;
typedef __attribute__((ext_vector_type(8))) int v8i;

#if defined(__has_builtin)
#if __has_builtin(__builtin_amdgcn_tensor_load_to_lds) && __has_builtin(__builtin_amdgcn_s_wait_tensorcnt)
#define HAVE_TDM 1
#endif
#endif

struct alignas(16) WaveScr {
  float Sig[1024];   // covariance carry
  float MH[1024];    // H / G / J
  float MA[1024];
  float MB[1024];
  float MC[1024];
  float MD[1024];
  float vec[384];    // mu | muz | rvec | tvec | hrelu[128] | hmask[128]
};

#if HAVE_TDM
// TDM DMA: copy nElem f32 (nElem <= 65535) from global to LDS.
// D# per CDNA5 ISA 8.3/8.4: 1-row tile, data_size=4B, tile_dim0=tensor_dim0=nElem.
__device__ inline void tdm_load_f32(float* lds_dst, const float* gsrc, int nElem) {
  unsigned long long ga = (unsigned long long)(uintptr_t)gsrc;
  unsigned int lo = (unsigned int)(uintptr_t)lds_dst;   // low 32 bits = LDS byte offset
  v4u g0;
  g0.x = 1u;                                             // count=1, user descriptor
  g0.y = lo;                                             // lds_addr
  g0.z = (unsigned int)ga;                               // global_addr[31:0]
  g0.w = (unsigned int)((ga >> 32) & 0x01FFFFFFull) | 0x80000000u; // addr[56:32] | type=2
  v8i g1;
  g1[0] = 0x00020000;                                    // data_size=2 (4 bytes)
  g1[1] = (nElem & 0xFFFF) << 16;                        // tensor_dim0[15:0]
  g1[2] = ((nElem >> 16) & 0xFFFF) | (1 << 16);          // tensor_dim0[31:16] | tensor_dim1=1
  g1[3] = (nElem & 0xFFFF) << 16;                        // tile_dim0
  g1[4] = 1;                                             // tile_dim1=1, tile_dim2=0
  g1[5] = nElem;                                         // tensor_dim0_stride[31:0]
  g1[6] = ((nElem >> 16) & 0xFFFF) | ((nElem & 0xFFFF) << 16); // stride0 hi | stride1 lo
  g1[7] = (nElem >> 16) & 0xFFFF;                        // tensor_dim1_stride hi
  v4i z4 = {};
#if __clang_major__ >= 23
  v8i z8 = {};
  __builtin_amdgcn_tensor_load_to_lds(g0, g1, z4, z4, z8, 0);
#else
  __builtin_amdgcn_tensor_load_to_lds(g0, g1, z4, z4, 0);
#endif
}
#endif

// C(32x32) = alpha * A(32xK) * op(B)  [+ diagAdd*I] [+ C if accumC]
// A optionally column-scaled by maskA[k] (for W2 * diag(mask) * W1 jacobians).
// Fragment layouts per CDNA5 ISA 7.12.2 (f32, 16x16x4).
__device__ inline void wave_mm(const float* A, int lda, const float* B, int ldb,
                               float* C, int ldc, int K, bool transB,
                               float alpha, float diagAdd, bool accumC,
                               const float* maskA) {
  const int lane = threadIdx.x & 31;
  const int lq   = lane & 15;
  const int half = lane >> 4;
  __syncthreads();
  for (int mt = 0; mt < 32; mt += 16) {
    for (int nt = 0; nt < 32; nt += 16) {
      v8f acc = {};
      for (int kt = 0; kt < K; kt += 4) {
        const int k0 = kt + 2 * half;
        v2f a, bf;
        float am0 = maskA ? maskA[k0]     : 1.0f;
        float am1 = maskA ? maskA[k0 + 1] : 1.0f;
        a.x = A[(mt + lq) * lda + k0]     * am0;
        a.y = A[(mt + lq) * lda + k0 + 1] * am1;
        if (transB) {
          bf.x = B[(nt + lq) * ldb + k0];
          bf.y = B[(nt + lq) * ldb + k0 + 1];
        } else {
          bf.x = B[(k0)     * ldb + nt + lq];
          bf.y = B[(k0 + 1) * ldb + nt + lq];
        }
        acc = __builtin_amdgcn_wmma_f32_16x16x4_f32(false, a, false, bf,
                                                    (short)0, acc, false, false);
      }
#pragma unroll
      for (int v = 0; v < 8; v++) {
        const int m = mt + v + 8 * half;
        const int n = nt + lq;
        float r = alpha * acc[v];
        if (m == n) r += diagAdd;
        if (accumC) r += C[m * ldc + n];
        C[m * ldc + n] = r;
      }
    }
  }
  __syncthreads();
}

// In-place Gauss-Jordan inverse of a 32x32 LDS matrix (no pivoting; S is SPD+R).
// Lane j owns column j. volatile pins LDS program order (wave-internal, in-order).
__device__ inline void wave_inv32(float* S) {
  volatile float* Sv = S;
  const int j = threadIdx.x & 31;
  __syncthreads();
#pragma unroll 1
  for (int k = 0; k < 32; k++) {
    const float pinv = 1.0f / Sv[k * 32 + k];
    const float rk   = Sv[k * 32 + j];
#pragma unroll 1
    for (int i = 0; i < 32; i++) {
      if (i == k) continue;
      const float f   = Sv[i * 32 + k];
      const float cur = Sv[i * 32 + j];
      Sv[i * 32 + j] = (j == k) ? (-f * pinv) : (cur - f * (rk * pinv));
    }
    Sv[k * 32 + j] = (j == k) ? pinv : rk * pinv;
  }
  __syncthreads();
}

// 32x32 matrix LDS -> global, float4-vectorized (b128 traffic).
__device__ inline void wave_mat_store(float* __restrict__ g, const float* l) {
  const int lane = threadIdx.x & 31;
  __syncthreads();
  const float4* ls = (const float4*)l;
  float4* gs = (float4*)g;
#pragma unroll
  for (int rr = 0; rr < 8; rr++) gs[rr * 32 + lane] = ls[rr * 32 + lane];
}

// 32x32 matrix global -> LDS.
__device__ inline void wave_mat_load(float* l, const float* __restrict__ g) {
  const int lane = threadIdx.x & 31;
  float4* ld = (float4*)l;
  const float4* gsrc = (const float4*)g;
#pragma unroll
  for (int rr = 0; rr < 8; rr++) ld[rr * 32 + lane] = gsrc[rr * 32 + lane];
  __syncthreads();
}

// MLP forward: h=W1@x+b1 (relu+mask kept), y=W2@relu(h)+b2.  x,yout in LDS (32).
__device__ inline void wave_mlp(const float* x, const float* W1, const float* b1,
                                const float* W2, const float* b2,
                                float* yout, float* hrelu, float* hmask) {
  const int lane = threadIdx.x & 31;
  __syncthreads();
#pragma unroll
  for (int rr = 0; rr < 4; rr++) {
    const int h = lane + 32 * rr;
    float acc = b1[h];
#pragma unroll
    for (int jj = 0; jj < LATENT; jj++) acc += W1[h * LATENT + jj] * x[jj];
    hrelu[h] = fmaxf(acc, 0.0f);
    hmask[h] = acc > 0.0f ? 1.0f : 0.0f;
  }
  __syncthreads();
  float acc = b2[lane];
#pragma unroll 4
  for (int hh = 0; hh < HIDN; hh++) acc += W2[lane * HIDN + hh] * hrelu[hh];
  yout[lane] = acc;
  __syncthreads();
}

__global__ __launch_bounds__(NTHREADS, 1)
void ekf_vae_wmma_kernel(const float* __restrict__ obs,
                         const float* __restrict__ tW1, const float* __restrict__ tb1,
                         const float* __restrict__ tW2, const float* __restrict__ tb2,
                         const float* __restrict__ eW1, const float* __restrict__ eb1,
                         const float* __restrict__ eW2, const float* __restrict__ eb2,
                         float* __restrict__ out, float* __restrict__ ws) {
  __shared__ float s_tW1[HIDN * LATENT];
  __shared__ float s_tW2[LATENT * HIDN];
  __shared__ float s_eW1[HIDN * LATENT];
  __shared__ float s_eW2[OBSN * HIDN];
  __shared__ float s_tb1[HIDN], s_tb2[LATENT], s_eb1[HIDN], s_eb2[OBSN];
  __shared__ WaveScr scr[WPB];

  const int tid  = threadIdx.x;
  const int wv   = tid >> 5;
  const int lane = tid & 31;
  const int b    = blockIdx.x * WPB + wv;

#if HAVE_TDM
  // Stage the four 4096-element weight matrices with the Tensor Data Mover.
  if (wv == 0) {
    tdm_load_f32(s_tW1, tW1, HIDN * LATENT);
    tdm_load_f32(s_tW2, tW2, LATENT * HIDN);
    tdm_load_f32(s_eW1, eW1, HIDN * LATENT);
    tdm_load_f32(s_eW2, eW2, OBSN * HIDN);
    __builtin_amdgcn_s_wait_tensorcnt(0);
  }
#else
  for (int i = tid; i < HIDN * LATENT; i += NTHREADS) {
    s_tW1[i] = tW1[i]; s_eW1[i] = eW1[i];
    s_tW2[i] = tW2[i]; s_eW2[i] = eW2[i];
  }
#endif
  for (int i = tid; i < HIDN; i += NTHREADS) { s_tb1[i] = tb1[i]; s_eb1[i] = eb1[i]; }
  for (int i = tid; i < LATENT; i += NTHREADS) { s_tb2[i] = tb2[i]; s_eb2[i] = eb2[i]; }
  __syncthreads();

  const size_t TB = (size_t)TSTEPS * BATCH;
  float* out_mu  = out;                 // (T,B,32): mu_filt during fwd, mu_smooth after bwd
  float* out_Sig = out + TB * LATENT;   // (T,B,32,32): Sig_filt during fwd, Sig_smooth after
  float* ws_G    = ws;                  // (T,B,32,32)

  WaveScr& S = scr[wv];
  float* Sig = S.Sig; float* MH = S.MH; float* MA = S.MA;
  float* MB  = S.MB;  float* MC = S.MC; float* MD = S.MD;
  float* mu    = S.vec;        float* muz   = S.vec + 32;
  float* rvec  = S.vec + 64;   float* tvec  = S.vec + 96;
  float* hrelu = S.vec + 128;  float* hmask = S.vec + 256;

  // init carry: mu0 = 0, Sig0 = 20*I
  mu[lane] = 0.0f;
#pragma unroll
  for (int r = 0; r < 32; r++) Sig[r * 32 + lane] = (r == lane) ? 20.0f : 0.0f;
  __syncthreads();

  // ------------------------- forward filter -------------------------
#pragma unroll 1
  for (int t = 0; t < TSTEPS; t++) {
    const size_t tb = (size_t)t * BATCH + b;
    if (t + 1 < TSTEPS)  // hide next observation's latency (global_prefetch_b8)
      __builtin_prefetch(&obs[((size_t)b * TSTEPS + t + 1) * OBSN], 0, 1);
    // encoder MLP + jacobian H = eW2 * diag(mask) * eW1
    wave_mlp(mu, s_eW1, s_eb1, s_eW2, s_eb2, tvec, hrelu, hmask);
    rvec[lane] = obs[((size_t)b * TSTEPS + t) * OBSN + lane] - tvec[lane];
    __syncthreads();
    wave_mm(s_eW2, HIDN, s_eW1, LATENT, MH, 32, HIDN, false, 1.0f, 0.0f, false, hmask);
    wave_mm(Sig, 32, MH, 32, MA, 32, 32, true,  1.0f, 0.0f,  false, nullptr);  // SHt = Sig*H^T
    wave_mm(MH,  32, MA, 32, MB, 32, 32, false, 1.0f, 0.03f, false, nullptr);  // S = H*SHt + R
    wave_inv32(MB);                                                            // S^-1
    wave_mm(MA,  32, MB, 32, MC, 32, 32, false, 1.0f, 0.0f,  false, nullptr);  // K = SHt*Sinv
    {
      float acc = mu[lane];
#pragma unroll
      for (int jj = 0; jj < 32; jj++) acc += MC[lane * 32 + jj] * rvec[jj];
      muz[lane] = acc;                                                         // mu_z = mu + K*r
    }
    __syncthreads();
    wave_mm(MC, 32, MH,  32, MA, 32, 32, false, -1.0f, 1.0f, false, nullptr);  // IKH = I - K*H
    wave_mm(MA, 32, Sig, 32, MB, 32, 32, false,  1.0f, 0.0f, false, nullptr);  // IKH*Sig
    wave_mm(MB, 32, MA,  32, MD, 32, 32, true,   1.0f, 0.0f, false, nullptr);  // ..*IKH^T
    wave_mm(MC, 32, MC,  32, MD, 32, 32, true,   0.03f, 0.0f, true, nullptr);  // += K*R*K^T
    // park filtered state in out (smoother rewrites t <= T-2 in place later)
    out_mu[tb * LATENT + lane] = muz[lane];
    wave_mat_store(out_Sig + tb * 1024, MD);
    // transition MLP + jacobian G
    wave_mlp(muz, s_tW1, s_tb1, s_tW2, s_tb2, tvec, hrelu, hmask);
    wave_mm(s_tW2, HIDN, s_tW1, LATENT, MH, 32, HIDN, false, 1.0f, 0.0f, false, hmask);
    wave_mat_store(ws_G + tb * 1024, MH);
    wave_mm(MH, 32, MD, 32, MA, 32, 32, false, 1.0f, 0.0f,  false, nullptr);   // G*Sig_z
    wave_mm(MA, 32, MH, 32, Sig, 32, 32, true, 1.0f, 0.08f, false, nullptr);   // + Q -> carry
    mu[lane] = tvec[lane];
    __syncthreads();
  }

  // ------------------------- RTS smoother (reverse) -------------------------
  {
    // carry = (mu_filt[T-1], Sig_filt[T-1]); t = T-1 output already in place.
    const size_t tbL = (size_t)(TSTEPS - 1) * BATCH + b;
    wave_mat_load(Sig, out_Sig + tbL * 1024);
    mu[lane] = out_mu[tbL * LATENT + lane];
    __syncthreads();
  }
#pragma unroll 1
  for (int t = TSTEPS - 2; t >= 0; t--) {
    const size_t tb = (size_t)t * BATCH + b;
    // mu_pred[t+1] = t_MLP(mu_filt[t])  (recomputed; saves workspace)
    tvec[lane] = out_mu[tb * LATENT + lane];   // mu_filt[t]
    __syncthreads();
    wave_mlp(tvec, s_tW1, s_tb1, s_tW2, s_tb2, rvec, hrelu, hmask);
    rvec[lane] = mu[lane] - rvec[lane];        // dvec = mu_s - mu_pred[t+1]
    __syncthreads();
    wave_mat_load(MA, out_Sig + tb * 1024);    // Sf = Sig_filt[t]
    wave_mat_load(MH, ws_G + tb * 1024);       // G[t]
    // Sig_pred[t+1] = G*Sf*G^T + Q  (recomputed)
    wave_mm(MH, 32, MA, 32, MC, 32, 32, false, 1.0f, 0.0f,  false, nullptr);
    wave_mm(MC, 32, MH, 32, MB, 32, 32, true,  1.0f, 0.08f, false, nullptr);
#pragma unroll
    for (int r = 0; r < 32; r++) MD[r * 32 + lane] = Sig[r * 32 + lane] - MB[r * 32 + lane];
    __syncthreads();
    wave_inv32(MB);                                                           // inv(Sig_pred)
    wave_mm(MA, 32, MH, 32, MC, 32, 32, true,  1.0f, 0.0f, false, nullptr);   // Sf*G^T
    wave_mm(MC, 32, MB, 32, MH, 32, 32, false, 1.0f, 0.0f, false, nullptr);   // J
    {
      float acc = tvec[lane];                  // mu_filt[t]
#pragma unroll
      for (int jj = 0; jj < 32; jj++) acc += MH[lane * 32 + jj] * rvec[jj];
      mu[lane] = acc;                          // mu_s
    }
    __syncthreads();
    wave_mm(MH, 32, MD, 32, MC, 32, 32, false, 1.0f, 0.0f, false, nullptr);   // J*(Ss-Sp)
#pragma unroll
    for (int r = 0; r < 32; r++) Sig[r * 32 + lane] = MA[r * 32 + lane];      // Sig_s = Sf ..
    __syncthreads();
    wave_mm(MC, 32, MH, 32, Sig, 32, 32, true, 1.0f, 0.0f, true, nullptr);    // .. + (JD)J^T
    out_mu[tb * LATENT + lane] = mu[lane];
    wave_mat_store(out_Sig + tb * 1024, Sig);
  }
}

extern "C" void kernel_launch(void* const* d_in, const int* in_sizes, int n_in,
                              void* d_out, int out_size, void* d_ws, size_t ws_size,
                              hipStream_t stream) {
  (void)in_sizes; (void)n_in; (void)out_size; (void)ws_size;
  const float* obs = (const float*)d_in[0];
  const float* tW1 = (const float*)d_in[1];
  const float* tb1 = (const float*)d_in[2];
  const float* tW2 = (const float*)d_in[3];
  const float* tb2 = (const float*)d_in[4];
  const float* eW1 = (const float*)d_in[5];
  const float* eb1 = (const float*)d_in[6];
  const float* eW2 = (const float*)d_in[7];
  const float* eb2 = (const float*)d_in[8];
  dim3 grid(BATCH / WPB);
  dim3 block(NTHREADS);
  ekf_vae_wmma_kernel<<<grid, block, 0, stream>>>(obs, tW1, tb1, tW2, tb2,
                                                  eW1, eb1, eW2, eb2,
                                                  (float*)d_out, (float*)d_ws);
}